// NeuralAutomaton_19731079758573
// MI455X (gfx1250) — compile-verified
//
#include <hip/hip_runtime.h>

// ---------------------------------------------------------------------------
// Neural automaton: final_state = (Π_t P_t) @ start,  P_t = Σ_s x[t,s]·A_s
// Strategy: chunked associative matrix product with f32 WMMA (16x16x4),
// then a tiny sequential matvec reduction over chunk matrices.
// ---------------------------------------------------------------------------

#define NUM_STATES   128
#define NUM_SYMBOLS  32
#define NUM_CLASSES  16
#define T_STATES     144            // 128 + 16
#define SEQ_LEN      8192
#define CHUNK_L      64
#define NUM_CHUNKS   (SEQ_LEN / CHUNK_L)   // 128

#define P_ROWS   128                 // bottom 16 rows of every P_t are zero
#define PS       148                 // padded LDS row stride (floats), conflict-free
#define MS       148
#define KTILES   (T_STATES / 4)      // 36 k-steps of 4 per WMMA chain

#define P_ELEMS  (P_ROWS * PS)                   // 18944 floats
#define M_ELEMS  (T_STATES * MS)                 // 21312 floats
#define X_ELEMS  (CHUNK_L * NUM_SYMBOLS)         // 2048 floats
#define LDS_FLOATS (P_ELEMS + 2 * M_ELEMS + X_ELEMS)   // 63616 floats = 248.5 KB

typedef float v2f __attribute__((ext_vector_type(2)));
typedef float v8f __attribute__((ext_vector_type(8)));

// ---------------------------------------------------------------------------
// Phase 1: each workgroup computes M_chunk = P_{t0+L-1} ... P_{t0}  (144x144)
// 288 threads = 9 wave32; wave w owns output column tile w (16 cols).
// Register blocking: all 8 row-tile accumulators (64 VGPRs) stay live;
// each B fragment is loaded once per k-step and reused by 8 WMMAs.
// ---------------------------------------------------------------------------
__global__ __launch_bounds__(288, 1)
void na_chunk_kernel(const float* __restrict__ x,
                     const float* __restrict__ totalP,
                     float* __restrict__ chunkM)
{
    extern __shared__ float lds[];
    float* Pg    = lds;                       // P_t (128 x 144, padded)
    float* Mbuf0 = lds + P_ELEMS;             // ping
    float* Mbuf1 = Mbuf0 + M_ELEMS;           // pong
    float* xs    = Mbuf1 + M_ELEMS;           // x rows for this chunk

    const int tid   = threadIdx.x;
    const int chunk = blockIdx.x;
    const int lane  = tid & 31;
    const int wave  = tid >> 5;               // 0..8
    const int half  = lane >> 4;              // 0/1  (lane[4])
    const int l15   = lane & 15;

    // stage this chunk's x rows into LDS
    for (int i = tid; i < X_ELEMS; i += blockDim.x)
        xs[i] = x[chunk * X_ELEMS + i];

    // M0 = Identity
    for (int i = tid; i < T_STATES * T_STATES; i += blockDim.x) {
        int r = i / T_STATES, c = i - r * T_STATES;
        Mbuf0[r * MS + c] = (r == c) ? 1.0f : 0.0f;
    }

    float* Mcur = Mbuf0;
    float* Mnxt = Mbuf1;

    const int ct    = wave;                   // column tile 0..8
    const int cbase = ct * 16 + l15;          // this lane's output column

    __syncthreads();

    for (int t = 0; t < CHUNK_L; ++t) {
        // ---- hoist this timestep's 32 symbol weights into registers
        float xw[NUM_SYMBOLS];
        #pragma unroll
        for (int s = 0; s < NUM_SYMBOLS; ++s)
            xw[s] = xs[t * NUM_SYMBOLS + s];

        // ---- build P_t (top 128 rows): P[n][m] = sum_s xw[s] * A[s][n][m]
        for (int g = tid; g < P_ROWS * (T_STATES / 4); g += blockDim.x) {
            int n  = g / (T_STATES / 4);
            int m4 = (g - n * (T_STATES / 4)) * 4;
            const float* ap = totalP + n * T_STATES + m4;
            float4 acc4 = make_float4(0.f, 0.f, 0.f, 0.f);
            #pragma unroll 8
            for (int s = 0; s < NUM_SYMBOLS; ++s) {
                float  w = xw[s];
                float4 a = *(const float4*)(ap + s * (T_STATES * T_STATES));
                acc4.x += w * a.x; acc4.y += w * a.y;
                acc4.z += w * a.z; acc4.w += w * a.w;
            }
            float* pp = &Pg[n * PS + m4];
            pp[0] = acc4.x; pp[1] = acc4.y; pp[2] = acc4.z; pp[3] = acc4.w;
        }
        // bottom 16 rows of M_next are structurally zero (P's zero block)
        for (int i = tid; i < NUM_CLASSES * T_STATES; i += blockDim.x) {
            int r = NUM_STATES + i / T_STATES, c = i % T_STATES;
            Mnxt[r * MS + c] = 0.0f;
        }
        __syncthreads();

        // ---- M_next(top 128) = P_t @ M_cur  via v_wmma_f32_16x16x4_f32
        //      8 live accumulators; B loaded once per k-step, reused 8x.
        v8f acc[8];
        #pragma unroll
        for (int rt = 0; rt < 8; ++rt) acc[rt] = (v8f){0.f,0.f,0.f,0.f,0.f,0.f,0.f,0.f};

        const float* abase = &Pg[l15 * PS + half * 2];
        const float* bbase = &Mcur[(half * 2) * MS + cbase];

        #pragma unroll 4
        for (int kt = 0; kt < KTILES; ++kt) {
            v2f b;
            b.x = bbase[kt * 4 * MS];            // row k0,   this column
            b.y = bbase[kt * 4 * MS + MS];       // row k0+1, this column
            #pragma unroll
            for (int rt = 0; rt < 8; ++rt) {
                v2f a = *(const v2f*)(abase + rt * 16 * PS + kt * 4); // ds_load_b64
                acc[rt] = __builtin_amdgcn_wmma_f32_16x16x4_f32(
                              false, a, false, b, (short)0, acc[rt], false, false);
            }
        }

        #pragma unroll
        for (int rt = 0; rt < 8; ++rt) {
            #pragma unroll
            for (int i = 0; i < 8; ++i)
                Mnxt[(rt * 16 + half * 8 + i) * MS + cbase] = acc[rt][i];
        }
        __syncthreads();

        float* tmp = Mcur; Mcur = Mnxt; Mnxt = tmp;
    }

    // write chunk matrix (compact 144x144)
    float* out = chunkM + (size_t)chunk * (T_STATES * T_STATES);
    for (int i = tid; i < T_STATES * T_STATES; i += blockDim.x) {
        int r = i / T_STATES, c = i - r * T_STATES;
        out[i] = Mcur[r * MS + c];
    }
}

// ---------------------------------------------------------------------------
// Phase 2: state = M_{C-1} ... M_0 @ start ; emit class probabilities.
// One workgroup, 128 sequential 144x144 matvecs (tiny).
// ---------------------------------------------------------------------------
__global__ __launch_bounds__(256)
void na_reduce_kernel(const float* __restrict__ chunkM,
                      const float* __restrict__ start,
                      float* __restrict__ out)
{
    __shared__ float st[T_STATES];
    const int tid = threadIdx.x;
    if (tid < T_STATES) st[tid] = start[tid];
    __syncthreads();

    for (int c = 0; c < NUM_CHUNKS; ++c) {
        float acc = 0.0f;
        if (tid < T_STATES) {
            const float* Mrow = chunkM + (size_t)c * T_STATES * T_STATES
                                       + (size_t)tid * T_STATES;
            #pragma unroll 4
            for (int m = 0; m < T_STATES; ++m)
                acc += Mrow[m] * st[m];
        }
        __syncthreads();
        if (tid < T_STATES) st[tid] = acc;
        __syncthreads();
    }

    if (tid >= NUM_STATES && tid < T_STATES)
        out[tid - NUM_STATES] = st[tid];
}

// ---------------------------------------------------------------------------
extern "C" void kernel_launch(void* const* d_in, const int* in_sizes, int n_in,
                              void* d_out, int out_size, void* d_ws, size_t ws_size,
                              hipStream_t stream)
{
    const float* x      = (const float*)d_in[0];   // [8192, 32]
    const float* totalP = (const float*)d_in[1];   // [32, 144, 144]
    const float* start  = (const float*)d_in[2];   // [144]
    float*       outp   = (float*)d_out;           // [16]
    float*       chunkM = (float*)d_ws;            // 128 * 144*144 f32 = 10.6 MB

    const size_t ldsBytes = (size_t)LDS_FLOATS * sizeof(float);
    na_chunk_kernel<<<NUM_CHUNKS, 288, ldsBytes, stream>>>(x, totalP, chunkM);
    na_reduce_kernel<<<1, 256, 0, stream>>>(chunkM, start, outp);
}